// LBPModule_81922206204551
// MI455X (gfx1250) — compile-verified
//
#include <hip/hip_runtime.h>
#include <hip/hip_bf16.h>

typedef __attribute__((ext_vector_type(2))) float v2f;
typedef __attribute__((ext_vector_type(8))) float v8f;

#define Bsz 8
#define Csz 64
#define Hsz 256
#define Wsz 256
#define HW  (Hsz * Wsz)          // 65536
#define HP  (Hsz + 1)            // 257
#define WP  (Wsz + 1)            // 257
#define NP  (HP * WP)            // 66049

// -----------------------------------------------------------------------------
// Kernel 1: channel sum  s[b,h,w] = sum_c x[b,c,h,w]   via V_WMMA_F32_16X16X4_F32
// One wave handles 16 consecutive pixels; 16 accumulating WMMAs cover 64 channels.
// A (16x4 f32): lanes 0-15 -> M=0..15, VGPR0=K0,VGPR1=K1; lanes 16-31 -> K2,K3.
// B = ones (4x16), so D[m,n] = channel sum of pixel m, replicated over n.
// -----------------------------------------------------------------------------
__global__ void lbp_chansum_wmma(const float* __restrict__ x, float* __restrict__ s) {
    const int lane = threadIdx.x & 31;
    const int wib  = threadIdx.x >> 5;                       // 8 waves / block
    const int waveId = blockIdx.x * 8 + wib;
    const int pbase  = waveId * 16;                          // first pixel of group
    const int b  = pbase / HW;                               // HW % 16 == 0: no straddle
    const int hw = (pbase % HW) + (lane & 15);
    const float* __restrict__ xb = x + (size_t)b * Csz * HW;

    const int chi = (lane >> 4) * 2;                         // 0 for lanes 0-15, 2 for 16-31
    v2f bones; bones.x = 1.0f; bones.y = 1.0f;               // ones B-matrix (layout-invariant)
    v8f acc = {};

#pragma unroll
    for (int g = 0; g < 16; ++g) {
        const int c0 = 4 * g + chi;
        v2f a;
        a.x = xb[c0 * HW + hw];                              // K = chi
        a.y = xb[(c0 + 1) * HW + hw];                        // K = chi+1
        acc = __builtin_amdgcn_wmma_f32_16x16x4_f32(
            /*neg_a=*/false, a, /*neg_b=*/false, bones,
            /*c_mod=*/(short)0, acc, /*reuse_a=*/false, /*reuse_b=*/false);
    }

    // D[m,n] independent of n: lane 0 holds pixels 0..7 in acc[0..7],
    // lane 16 holds pixels 8..15. 64B-aligned stores (pbase % 16 == 0).
    float* out = s + pbase;
    if (lane == 0) {
        *(float4*)(out)     = make_float4(acc[0], acc[1], acc[2], acc[3]);
        *(float4*)(out + 4) = make_float4(acc[4], acc[5], acc[6], acc[7]);
    } else if (lane == 16) {
        *(float4*)(out + 8)  = make_float4(acc[0], acc[1], acc[2], acc[3]);
        *(float4*)(out + 12) = make_float4(acc[4], acc[5], acc[6], acc[7]);
    }
}

// -----------------------------------------------------------------------------
// Kernel 2: 2x2 stride-1 avg pool with 1-pad (count_include_pad): xs[b,i,j],
// window covers original rows i-1..i, cols j-1..j; OOB counts as zero; /4.
// -----------------------------------------------------------------------------
__global__ void lbp_pool(const float* __restrict__ s, float* __restrict__ xs) {
    const int idx = blockIdx.x * blockDim.x + threadIdx.x;
    if (idx >= Bsz * NP) return;
    const int b = idx / NP;
    const int i = (idx / WP) % HP;
    const int j = idx % WP;
    const float* __restrict__ sb = s + b * HW;
    float a = 0.0f;
#pragma unroll
    for (int dr = -1; dr <= 0; ++dr)
#pragma unroll
        for (int dc = -1; dc <= 0; ++dc) {
            const int r = i + dr, c = j + dc;
            if (r >= 0 && r < Hsz && c >= 0 && c < Wsz) a += sb[r * Wsz + c];
        }
    xs[idx] = a * 0.25f;
}

// -----------------------------------------------------------------------------
// Kernel 3: shift-difference conv.  Index pairs for K=5 are point-symmetric:
// feats[m] = xs(+off_m) - xs(-off_m); y = sum_m w[m]*feats[m] + bias.
// -----------------------------------------------------------------------------
__global__ void lbp_conv(const float* __restrict__ xs, const float* __restrict__ w,
                         const float* __restrict__ bias, float* __restrict__ y) {
    const int idx = blockIdx.x * blockDim.x + threadIdx.x;
    if (idx >= Bsz * NP) return;
    const int b = idx / NP;
    const int i = (idx / WP) % HP;
    const int j = idx % WP;
    const float* __restrict__ xb = xs + b * NP;

    const int RO[8] = { 2,  2,  2,  2,  2,  1,  0, -1};
    const int CO[8] = { 2,  1,  0, -1, -2, -2, -2, -2};

    float a = bias[0];
#pragma unroll
    for (int m = 0; m < 8; ++m) {
        const int rp = i + RO[m], cp = j + CO[m];
        const int rn = i - RO[m], cn = j - CO[m];
        const float vp = (rp >= 0 && rp < HP && cp >= 0 && cp < WP) ? xb[rp * WP + cp] : 0.0f;
        const float vn = (rn >= 0 && rn < HP && cn >= 0 && cn < WP) ? xb[rn * WP + cn] : 0.0f;
        a += w[m] * (vp - vn);
    }
    y[idx] = a;
}

// -----------------------------------------------------------------------------
// Kernel 4: per-image LayerNorm (no affine) over 257*257 + leaky ReLU, in place.
// One block per batch image; LDS tree reduction for sum / sumsq.
// -----------------------------------------------------------------------------
__global__ void lbp_ln_lrelu(float* __restrict__ y) {
    const int b = blockIdx.x;
    float* __restrict__ yb = y + b * NP;

    float s0 = 0.0f, s1 = 0.0f;
    for (int i = threadIdx.x; i < NP; i += blockDim.x) {
        const float v = yb[i];
        s0 += v; s1 += v * v;
    }

    __shared__ float rs[256];
    __shared__ float rq[256];
    rs[threadIdx.x] = s0; rq[threadIdx.x] = s1;
    __syncthreads();
    for (int off = 128; off > 0; off >>= 1) {
        if (threadIdx.x < (unsigned)off) {
            rs[threadIdx.x] += rs[threadIdx.x + off];
            rq[threadIdx.x] += rq[threadIdx.x + off];
        }
        __syncthreads();
    }

    __shared__ float mu_s, rsig_s;
    if (threadIdx.x == 0) {
        const float inv = 1.0f / (float)NP;
        const float mu  = rs[0] * inv;
        const float var = fmaxf(rq[0] * inv - mu * mu, 0.0f);
        mu_s   = mu;
        rsig_s = rsqrtf(var + 1e-5f);
    }
    __syncthreads();

    const float mu = mu_s, rsig = rsig_s;
    for (int i = threadIdx.x; i < NP; i += blockDim.x) {
        const float v = (yb[i] - mu) * rsig;
        yb[i] = v >= 0.0f ? v : 0.01f * v;
    }
}

// -----------------------------------------------------------------------------
extern "C" void kernel_launch(void* const* d_in, const int* in_sizes, int n_in,
                              void* d_out, int out_size, void* d_ws, size_t ws_size,
                              hipStream_t stream) {
    const float* x      = (const float*)d_in[0];   // [8,64,256,256]
    const float* conv_w = (const float*)d_in[1];   // [1,8]
    const float* conv_b = (const float*)d_in[2];   // [1]
    float* out = (float*)d_out;                    // [8,1,257,257]

    float* s  = (float*)d_ws;                      // [8,256,256]  = 2.00 MB
    float* xs = s + (size_t)Bsz * HW;              // [8,257,257]  = 2.11 MB

    // 1) channel sum via WMMA: 524288 pixels / 16 per wave / 8 waves per block
    lbp_chansum_wmma<<<(Bsz * HW) / (16 * 8), 256, 0, stream>>>(x, s);

    // 2-3) pool, shift-conv (tiny, L2 resident)
    const int n2 = Bsz * NP;
    lbp_pool<<<(n2 + 255) / 256, 256, 0, stream>>>(s, xs);
    lbp_conv<<<(n2 + 255) / 256, 256, 0, stream>>>(xs, conv_w, conv_b, out);

    // 4) per-image layernorm + leaky relu, in place on d_out
    lbp_ln_lrelu<<<Bsz, 256, 0, stream>>>(out);
}